// BilateralFilter_76605036691896
// MI455X (gfx1250) — compile-verified
//
#include <hip/hip_runtime.h>
#include <hip/hip_bf16.h>

// Bilateral 5x5 filter, edge-clamped, for x:[B,C,512,512] f32.
// CDNA5 strategy: async global->LDS tile staging (ASYNCcnt path),
// exp via v_exp_f32 (exp2 with folded constant), rcp instead of div.

#define KSIZE     5
#define PAD       2
#define TILE_W    64
#define TILE_H    16
#define LDS_W     (TILE_W + 2 * PAD)   // 68
#define LDS_H     (TILE_H + 2 * PAD)   // 20
#define LDS_N     (LDS_W * LDS_H)      // 1360 floats = 5440 B
#define NTHREADS  256                  // 8 wave32s

#define AS1 __attribute__((address_space(1)))
#define AS3 __attribute__((address_space(3)))

__device__ __forceinline__ void async_copy_f32_to_lds(const float* g, float* l) {
#if __has_builtin(__builtin_amdgcn_global_load_async_to_lds_b32)
    __builtin_amdgcn_global_load_async_to_lds_b32((AS1 int*)(void*)g,
                                                  (AS3 int*)(void*)l,
                                                  /*offset=*/0, /*cpol=*/0);
#else
    // GV mode: VDST = LDS byte offset (addr[31:0] of flat shared addr),
    // VADDR = 64-bit global address.
    unsigned lds_off = (unsigned)(unsigned long long)(void*)l;
    asm volatile("global_load_async_to_lds_b32 %0, %1, off"
                 :: "v"(lds_off), "v"(g) : "memory");
#endif
}

__device__ __forceinline__ void wait_async_zero() {
#if __has_builtin(__builtin_amdgcn_s_wait_asynccnt)
    __builtin_amdgcn_s_wait_asynccnt(0);
#else
    asm volatile("s_wait_asynccnt 0" ::: "memory");
#endif
}

__device__ __forceinline__ float fast_exp2(float x) {
#if __has_builtin(__builtin_amdgcn_exp2f)
    return __builtin_amdgcn_exp2f(x);
#else
    return exp2f(x);
#endif
}

__global__ __launch_bounds__(NTHREADS)
void BilateralFilter_76605036691896_kernel(const float* __restrict__ x,
                                           const float* __restrict__ gw,
                                           float* __restrict__ out,
                                           int H, int W) {
    __shared__ float tile[LDS_N];

    const int tid = threadIdx.x;
    const int bc  = blockIdx.z;
    const int ty0 = blockIdx.y * TILE_H;
    const int tx0 = blockIdx.x * TILE_W;

    const float* __restrict__ xp = x + (size_t)bc * H * W;

    // ---- Stage (TILE+halo) into LDS via async copies, edge-clamped ----
    for (int idx = tid; idx < LDS_N; idx += NTHREADS) {
        const int ty = idx / LDS_W;
        const int tx = idx - ty * LDS_W;
        int gy = ty0 - PAD + ty;
        int gx = tx0 - PAD + tx;
        gy = gy < 0 ? 0 : (gy > H - 1 ? H - 1 : gy);
        gx = gx < 0 ? 0 : (gx > W - 1 ? W - 1 : gx);
        async_copy_f32_to_lds(xp + (size_t)gy * W + gx, &tile[idx]);
    }
    wait_async_zero();   // this wave's async copies done
    __syncthreads();     // all waves' LDS writes visible

    // ---- Spatial gaussian weights: uniform -> scalar loads into SGPRs ----
    float g[KSIZE * KSIZE];
#pragma unroll
    for (int k = 0; k < KSIZE * KSIZE; ++k) g[k] = gw[k];

    // exp(-d^2 / (2*sigma^2)) == exp2(d^2 * K2), K2 = -log2(e)/(2*sigma^2)
    const float SIGMA = 0.3f * ((KSIZE - 1) * 0.5f - 1.0f) + 0.8f;  // 1.1
    const float K2 = -1.44269504088896340736f / (2.0f * SIGMA * SIGMA);

    const int lx  = tid & (TILE_W - 1);   // 0..63
    const int lyb = tid >> 6;             // 0..3

#pragma unroll
    for (int r = 0; r < 4; ++r) {
        const int ly = lyb + r * 4;       // 0..15
        const float c = tile[(ly + PAD) * LDS_W + (lx + PAD)];
        float num = 0.0f;
        float den = 0.0f;
#pragma unroll
        for (int i = 0; i < KSIZE; ++i) {
#pragma unroll
            for (int j = 0; j < KSIZE; ++j) {
                const float p = tile[(ly + i) * LDS_W + (lx + j)];
                const float d = p - c;
                const float w = g[i * KSIZE + j] * fast_exp2(d * d * K2);
                num = fmaf(w, p, num);
                den += w;
            }
        }
        const int oy = ty0 + ly;
        const int ox = tx0 + lx;
        if (oy < H && ox < W) {
#if __has_builtin(__builtin_amdgcn_rcpf)
            out[(size_t)bc * H * W + (size_t)oy * W + ox] =
                num * __builtin_amdgcn_rcpf(den);
#else
            out[(size_t)bc * H * W + (size_t)oy * W + ox] = num / den;
#endif
        }
    }
}

extern "C" void kernel_launch(void* const* d_in, const int* in_sizes, int n_in,
                              void* d_out, int out_size, void* d_ws, size_t ws_size,
                              hipStream_t stream) {
    const float* x  = (const float*)d_in[0];   // [B,C,512,512] f32
    const float* gw = (const float*)d_in[1];   // [5,5] f32
    float* out      = (float*)d_out;

    const int H = 512, W = 512;
    const int BC = in_sizes[0] / (H * W);      // B*C = 12

    dim3 grid(W / TILE_W, H / TILE_H, BC);     // (8, 32, 12) = 3072 blocks
    BilateralFilter_76605036691896_kernel<<<grid, NTHREADS, 0, stream>>>(
        x, gw, out, H, W);
}